// SOCNet_2370821947832
// MI455X (gfx1250) — compile-verified
//
#include <hip/hip_runtime.h>
#include <cstdint>
#include <cmath>

#define S_LEN   8192
#define TILE    128      // elements per tile per wave
#define VPT     4        // elements per lane (one float4 each)
#define NBUF    8        // LDS ring depth (async pipeline)
#define WPB     4        // waves per block
#define LANES   32

typedef float v4f __attribute__((ext_vector_type(4)));

__device__ __forceinline__ float softplus_f(float x) {
    // numerically stable softplus
    return fmaxf(x, 0.0f) + log1pf(expf(-fabsf(x)));
}

// Issue one tile (128 elems = 2KB) as 4 async B128 loads into LDS.
// The instruction offset is added to BOTH the global and the LDS address
// (CDNA5 ISA), so one (gbase,lbase) pair covers 64B/lane.
// Stream has zero reuse -> non-temporal hint keeps the 256MB X stream from
// churning L2.
__device__ __forceinline__ void issue_tile_async(uint64_t gbase, uint32_t lbase) {
    asm volatile("global_load_async_to_lds_b128 %0, %1, off th:TH_LOAD_NT"           :: "v"(lbase), "v"(gbase) : "memory");
    asm volatile("global_load_async_to_lds_b128 %0, %1, off offset:16 th:TH_LOAD_NT" :: "v"(lbase), "v"(gbase) : "memory");
    asm volatile("global_load_async_to_lds_b128 %0, %1, off offset:32 th:TH_LOAD_NT" :: "v"(lbase), "v"(gbase) : "memory");
    asm volatile("global_load_async_to_lds_b128 %0, %1, off offset:48 th:TH_LOAD_NT" :: "v"(lbase), "v"(gbase) : "memory");
}

extern "C" __global__ __launch_bounds__(WPB * LANES)
void soc_scan_kernel(const float* __restrict__ X,  const float* __restrict__ SC,
                     const float* __restrict__ W1, const float* __restrict__ b1,
                     const float* __restrict__ W2, const float* __restrict__ b2,
                     const float* __restrict__ Wa, const float* __restrict__ ba,
                     const float* __restrict__ Wb, const float* __restrict__ bb,
                     float* __restrict__ out, int B)
{
    __shared__ float4 lds_buf[WPB * NBUF * TILE];   // 64 KB per block

    const int wave = threadIdx.x >> 5;
    const int lane = threadIdx.x & 31;
    const int b    = blockIdx.x * WPB + wave;
    if (b >= B) return;                             // no barriers used -> safe

    const float4* Xrow = reinterpret_cast<const float4*>(X) + (size_t)b * S_LEN;
    float*        Orow = out + (size_t)b * S_LEN;

    // scalar parameters (broadcast loads, cached)
    const float w10 = W1[0], w11 = W1[1], w12 = W1[2], w13 = W1[3];
    const float b1v = b1[0], w2v = W2[0], b2v = b2[0];
    const float wa0 = Wa[0], wa1 = Wa[1], bav = ba[0];
    const float wbv = Wb[0], bbv = bb[0];
    const float Q    = SC[4 * b + 0];
    const float eta0 = SC[4 * b + 1];
    const float R    = SC[4 * b + 2];
    const float sc3  = SC[4 * b + 3];

    // SOC_init from the first sample of the row
    const float4 x0 = Xrow[0];
    const float h = softplus_f(w10 * x0.y + w11 * x0.z + w12 * x0.w + w13 * R + b1v);
    const float soc_init = sc3 * (1.0f + (w2v * h + b2v));
    const float coef = eta0 / (3600.0f * Q);

    float4* wbuf = lds_buf + wave * (NBUF * TILE);
    // generic shared pointer: low 32 bits == wave-relative LDS byte address
    const uint32_t wbase    = (uint32_t)(uintptr_t)(wbuf);
    const uint32_t lane_lds = (uint32_t)(lane * (VPT * 16));

    const int nt = S_LEN / TILE;                    // 64 tiles

    // prologue: fill the 8-deep ring (16 KB in flight per wave)
    #pragma unroll
    for (int p = 0; p < NBUF; ++p) {
        uint64_t g = (uint64_t)(uintptr_t)(Xrow + p * TILE + lane * VPT);
        uint32_t l = wbase + (uint32_t)(p * TILE * 16) + lane_lds;
        issue_tile_async(g, l);
    }

    float running = 0.0f;

    for (int i = 0; i < nt; ++i) {
        // Need tiles i and i+1 resident (i+1 supplies the boundary time).
        // Async loads complete in order, 4 instructions per tile:
        //  steady state: tiles i+2..i+NBUF-1 may be outstanding = 4*(NBUF-2) = 24.
        //  tail (all tiles issued): wait for everything (slightly conservative).
        if (i <= nt - NBUF) asm volatile("s_wait_asynccnt 0x18" ::: "memory");
        else                asm volatile("s_wait_asynccnt 0x0"  ::: "memory");

        const float4* buf = wbuf + (i & (NBUF - 1)) * TILE + lane * VPT;
        const float4 e0 = buf[0], e1 = buf[1], e2 = buf[2], e3 = buf[3];

        float t_next_tile = 0.0f;
        if (i + 1 < nt)
            t_next_tile = reinterpret_cast<const float*>(wbuf + ((i + 1) & (NBUF - 1)) * TILE)[0];

        // time of element (lane*4 + 4): lane+1's first t; lane 31 takes next tile's first t
        const float tnb = __shfl_down(e0.x, 1, LANES);
        const float t4  = (lane == LANES - 1) ? t_next_tile : tnb;

        // delta[s] = coef * (1 + Wb*softplus(Wa0*I + Wa1*T + ba) + bb) * I * dt
        const float g0 = coef * (1.0f + (wbv * softplus_f(wa0 * e0.y + wa1 * e0.z + bav) + bbv)) * e0.y;
        const float g1 = coef * (1.0f + (wbv * softplus_f(wa0 * e1.y + wa1 * e1.z + bav) + bbv)) * e1.y;
        const float g2 = coef * (1.0f + (wbv * softplus_f(wa0 * e2.y + wa1 * e2.z + bav) + bbv)) * e2.y;
        const float g3 = coef * (1.0f + (wbv * softplus_f(wa0 * e3.y + wa1 * e3.z + bav) + bbv)) * e3.y;

        const float d0 = g0 * (e1.x - e0.x);
        const float d1 = g1 * (e2.x - e1.x);
        const float d2 = g2 * (e3.x - e2.x);
        float       d3 = g3 * (t4   - e3.x);
        if (i == nt - 1 && lane == LANES - 1) d3 = 0.0f;   // s == S-1 has no delta

        const float lanesum = d0 + d1 + d2 + d3;

        // wave32 inclusive scan of per-lane sums (5 shuffle steps)
        float incl = lanesum;
        #pragma unroll
        for (int off = 1; off < LANES; off <<= 1) {
            float n = __shfl_up(incl, off, LANES);
            if (lane >= off) incl += n;
        }

        const float base = soc_init + running + (incl - lanesum);  // exclusive prefix

        v4f o;
        o.x = base;
        o.y = base + d0;
        o.z = base + d0 + d1;
        o.w = base + d0 + d1 + d2;
        // output is written once, never re-read by this kernel -> non-temporal
        __builtin_nontemporal_store(o, reinterpret_cast<v4f*>(Orow + (size_t)i * TILE + lane * VPT));

        running += __shfl(incl, LANES - 1, LANES);

        if (i + NBUF < nt) {
            uint64_t g = (uint64_t)(uintptr_t)(Xrow + (i + NBUF) * TILE + lane * VPT);
            uint32_t l = wbase + (uint32_t)(((i + NBUF) & (NBUF - 1)) * TILE * 16) + lane_lds;
            issue_tile_async(g, l);
        }
    }
}

extern "C" void kernel_launch(void* const* d_in, const int* in_sizes, int n_in,
                              void* d_out, int out_size, void* d_ws, size_t ws_size,
                              hipStream_t stream) {
    const float* X  = (const float*)d_in[0];
    const float* SC = (const float*)d_in[1];
    const float* W1 = (const float*)d_in[2];
    const float* b1 = (const float*)d_in[3];
    const float* W2 = (const float*)d_in[4];
    const float* b2 = (const float*)d_in[5];
    const float* Wa = (const float*)d_in[6];
    const float* ba = (const float*)d_in[7];
    const float* Wb = (const float*)d_in[8];
    const float* bb = (const float*)d_in[9];
    float* out = (float*)d_out;

    const int B = in_sizes[0] / (S_LEN * 4);        // X is (B, S, 4)
    dim3 grid((B + WPB - 1) / WPB);
    dim3 block(WPB * LANES);
    hipLaunchKernelGGL(soc_scan_kernel, grid, block, 0, stream,
                       X, SC, W1, b1, W2, b2, Wa, ba, Wb, bb, out, B);
}